// GraphClassifierExample_14353780704051
// MI455X (gfx1250) — compile-verified
//
#include <hip/hip_runtime.h>

typedef __attribute__((ext_vector_type(2))) float v2f;
typedef __attribute__((ext_vector_type(8))) float v8f;

#define DIM     128
#define N_GRAPH 64
#define N_CLS   10

// ---------------- utility ----------------
__global__ void fill_zero_kernel(float* __restrict__ p, int n) {
  int i = blockIdx.x * blockDim.x + threadIdx.x;
  if (i < n) p[i] = 0.0f;
}

// ---------------- degree counting ----------------
__global__ void degree_kernel(const int* __restrict__ src, const int* __restrict__ dst,
                              float* __restrict__ deg_out, float* __restrict__ deg_in,
                              int n_edges) {
  int e = blockIdx.x * blockDim.x + threadIdx.x;
  if (e < n_edges) {
    atomicAdd(&deg_out[src[e]], 1.0f);
    atomicAdd(&deg_in[dst[e]], 1.0f);
  }
}

// deg -> rsqrt(max(deg,1)) in place (applied to both deg arrays, contiguous)
__global__ void rsqrt_clip_kernel(float* __restrict__ p, int n) {
  int i = blockIdx.x * blockDim.x + threadIdx.x;
  if (i < n) {
    float v = p[i];
    v = v < 1.0f ? 1.0f : v;
    p[i] = rsqrtf(v);
  }
}

// ---------------- SpMM scatter: agg[dst] += h[src] * rout[src] ----------------
// One wave (32 lanes) per edge; each lane handles 4 contiguous channels (float4).
__global__ void spmm_scatter_kernel(const int* __restrict__ src, const int* __restrict__ dst,
                                    const float* __restrict__ h, const float* __restrict__ rout,
                                    float* __restrict__ agg, int n_edges) {
  int wave = (blockIdx.x * blockDim.x + threadIdx.x) >> 5;
  int lane = threadIdx.x & 31;
  if (wave >= n_edges) return;
  int s = src[wave];
  int d = dst[wave];
  float r = rout[s];
  const float4* hp = (const float4*)(h + (size_t)s * DIM);
  float4 v = hp[lane];                       // channels 4*lane .. 4*lane+3
  float* ap = agg + (size_t)d * DIM + lane * 4;
  atomicAdd(ap + 0, v.x * r);
  atomicAdd(ap + 1, v.y * r);
  atomicAdd(ap + 2, v.z * r);
  atomicAdd(ap + 3, v.w * r);
}

// ---------------- fused GEMM: out = relu((agg * rin) @ W + b) ----------------
// One wave computes one 16x16 tile of out. K=128 via 32 V_WMMA_F32_16X16X4_F32.
// A frag (16x4 f32): lane<16 -> K=k+0,k+1 ; lane>=16 -> K=k+2,k+3 ; row M = lane%16.
// B frag (4x16 f32): N = lane%16 ; vgpr0 -> K=k+2*(lane/16), vgpr1 -> K=+1.
// C/D (16x16 f32): N = lane%16, M = vgpr + 8*(lane/16).
__global__ void gcn_gemm_wmma_kernel(const float* __restrict__ A,    // [N, 128] agg
                                     const float* __restrict__ rin,  // [N] rsqrt(deg_in)
                                     const float* __restrict__ W,    // [128, 128] row-major
                                     const float* __restrict__ bias, // [128]
                                     float* __restrict__ out,        // [N, 128]
                                     int n_nodes) {
  int wave = (blockIdx.x * (blockDim.x >> 5)) + (threadIdx.x >> 5);
  int lane = threadIdx.x & 31;
  int n_mtiles = n_nodes >> 4;
  int mt = wave >> 3;          // 8 n-tiles per m-tile (128/16)
  int nt = wave & 7;
  if (mt >= n_mtiles) return;  // wave-uniform: EXEC stays all-ones for WMMA

  int m0 = mt << 4;
  int n0 = nt << 4;
  int l15 = lane & 15;
  int kg  = (lane >> 4) * 2;   // 0 or 2

  int   row = m0 + l15;
  float rs  = rin[row];
  const float* arow = A + (size_t)row * DIM + kg;            // pairs at k+kg
  const float* bcol = W + (size_t)kg * DIM + n0 + l15;       // rows k+kg, k+kg+1

  v8f c = {};
  for (int k = 0; k < DIM; k += 4) {
    v2f a;
    a.x = arow[k]     * rs;
    a.y = arow[k + 1] * rs;
    v2f b;
    b.x = bcol[(size_t)k * DIM];
    b.y = bcol[(size_t)k * DIM + DIM];
    c = __builtin_amdgcn_wmma_f32_16x16x4_f32(
        /*neg_a=*/false, a, /*neg_b=*/false, b,
        /*c_mod=*/(short)0, c, /*reuse_a=*/false, /*reuse_b=*/false);
  }

  int   nn = n0 + l15;
  float bv = bias[nn];
  int   mbase = m0 + (lane >> 4) * 8;
#pragma unroll
  for (int v = 0; v < 8; ++v) {
    float val = c[v] + bv;
    out[(size_t)(mbase + v) * DIM + nn] = val > 0.0f ? val : 0.0f;
  }
}

// ---------------- per-graph mean readout: accumulate ----------------
__global__ void readout_accum_kernel(const float* __restrict__ x, const int* __restrict__ gid,
                                     float* __restrict__ hg, float* __restrict__ cnt,
                                     int n_nodes) {
  int i = blockIdx.x * blockDim.x + threadIdx.x;   // node*128 + c
  if (i >= n_nodes * DIM) return;
  int node = i >> 7;
  int c = i & (DIM - 1);
  int g = gid[node];
  atomicAdd(&hg[(size_t)g * DIM + c], x[i]);
  if (c == 0) atomicAdd(&cnt[g], 1.0f);
}

// ---------------- classifier: out = (hg / cnt) @ Wc + bc ----------------
__global__ void classifier_kernel(const float* __restrict__ hg, const float* __restrict__ cnt,
                                  const float* __restrict__ Wc, const float* __restrict__ bc,
                                  float* __restrict__ out) {
  int i = blockIdx.x * blockDim.x + threadIdx.x;
  if (i >= N_GRAPH * N_CLS) return;
  int g = i / N_CLS;
  int cls = i - g * N_CLS;
  float cn = cnt[g];
  cn = cn < 1.0f ? 1.0f : cn;
  float inv = 1.0f / cn;
  float acc = 0.0f;
  const float* hrow = hg + (size_t)g * DIM;
#pragma unroll 4
  for (int k = 0; k < DIM; ++k)
    acc += hrow[k] * inv * Wc[(size_t)k * N_CLS + cls];
  out[i] = acc + bc[cls];
}

// ---------------- launch ----------------
extern "C" void kernel_launch(void* const* d_in, const int* in_sizes, int n_in,
                              void* d_out, int out_size, void* d_ws, size_t ws_size,
                              hipStream_t stream) {
  const float* h   = (const float*)d_in[0];
  const int*   src = (const int*)  d_in[1];
  const int*   dst = (const int*)  d_in[2];
  const int*   gid = (const int*)  d_in[3];
  const float* W1  = (const float*)d_in[4];
  const float* b1  = (const float*)d_in[5];
  const float* W2  = (const float*)d_in[6];
  const float* b2  = (const float*)d_in[7];
  const float* Wc  = (const float*)d_in[8];
  const float* bc  = (const float*)d_in[9];
  float* out = (float*)d_out;

  const int n_nodes = in_sizes[0] / DIM;   // 100000
  const int n_edges = in_sizes[1];         // 1600000
  const int nfeat   = n_nodes * DIM;       // 12,800,000

  // workspace layout (floats)
  float* agg  = (float*)d_ws;              // [n_nodes*128]
  float* xbuf = agg  + (size_t)nfeat;      // [n_nodes*128]
  float* rout = xbuf + (size_t)nfeat;      // [n_nodes]
  float* rin  = rout + n_nodes;            // [n_nodes]
  float* hg   = rin  + n_nodes;            // [64*128]
  float* cnt  = hg   + N_GRAPH * DIM;      // [64]

  const int B = 256;
  int small_n = 2 * n_nodes + N_GRAPH * DIM + N_GRAPH;  // rout..cnt contiguous

  // 1) zero scratch
  fill_zero_kernel<<<(nfeat + B - 1) / B, B, 0, stream>>>(agg, nfeat);
  fill_zero_kernel<<<(small_n + B - 1) / B, B, 0, stream>>>(rout, small_n);

  // 2) degrees -> rsqrt(clip)
  degree_kernel<<<(n_edges + B - 1) / B, B, 0, stream>>>(src, dst, rout, rin, n_edges);
  rsqrt_clip_kernel<<<(2 * n_nodes + B - 1) / B, B, 0, stream>>>(rout, 2 * n_nodes);

  const int spmm_blocks = (n_edges * 32 + B - 1) / B;          // wave per edge
  const int gemm_blocks = (n_nodes >> 4) * 8 / (B / 32);       // wave per 16x16 tile

  // 3) layer 1
  spmm_scatter_kernel<<<spmm_blocks, B, 0, stream>>>(src, dst, h, rout, agg, n_edges);
  gcn_gemm_wmma_kernel<<<gemm_blocks, B, 0, stream>>>(agg, rin, W1, b1, xbuf, n_nodes);

  // 4) layer 2
  fill_zero_kernel<<<(nfeat + B - 1) / B, B, 0, stream>>>(agg, nfeat);
  spmm_scatter_kernel<<<spmm_blocks, B, 0, stream>>>(src, dst, xbuf, rout, agg, n_edges);
  gcn_gemm_wmma_kernel<<<gemm_blocks, B, 0, stream>>>(agg, rin, W2, b2, xbuf, n_nodes);

  // 5) readout + classifier
  readout_accum_kernel<<<(nfeat + B - 1) / B, B, 0, stream>>>(xbuf, gid, hg, cnt, n_nodes);
  classifier_kernel<<<(N_GRAPH * N_CLS + B - 1) / B, B, 0, stream>>>(hg, cnt, Wc, bc, out);
}